// BaseParticleFilter_13288628814430
// MI455X (gfx1250) — compile-verified
//
#include <hip/hip_runtime.h>
#include <hip/hip_bf16.h>

#define NPART 65536
#define TSTEPS 256
#define DDIM 4
#define NBLK 256
#define TPB 256
#define RHO 0.95f
#define SIGX 0.5f
#define LOGN 11.0903548889591249f          /* ln(65536) */
#define OBSC 3.6757541328186907f           /* 0.5*4*ln(2*pi) */

typedef unsigned int u32;
typedef unsigned long long u64;
typedef u32 v4u __attribute__((ext_vector_type(4)));
typedef int  v8i __attribute__((ext_vector_type(8)));
typedef int  v4i __attribute__((ext_vector_type(4)));

#if defined(__HIP_DEVICE_COMPILE__) && __has_builtin(__builtin_amdgcn_tensor_load_to_lds)
#define HAVE_TDM 1
#else
#define HAVE_TDM 0
#endif

// ---------------- block reduction / scan helpers (256 threads) ----------------
__device__ __forceinline__ float blkSum(float v, float* sh) {
    int t = threadIdx.x;
    sh[t] = v; __syncthreads();
    for (int s = TPB / 2; s > 0; s >>= 1) {
        if (t < s) sh[t] = sh[t] + sh[t + s];
        __syncthreads();
    }
    float r = sh[0]; __syncthreads();
    return r;
}

__device__ __forceinline__ float blkMax(float v, float* sh) {
    int t = threadIdx.x;
    sh[t] = v; __syncthreads();
    for (int s = TPB / 2; s > 0; s >>= 1) {
        if (t < s) sh[t] = fmaxf(sh[t], sh[t + s]);
        __syncthreads();
    }
    float r = sh[0]; __syncthreads();
    return r;
}

// Combine 256 block partials into Z = logsumexp(lw) and Z2 = logsumexp(2*lw).
// Deterministic: every caller gets bitwise-identical results from the same arrays.
__device__ __forceinline__ void combineZZ2(const float* __restrict__ pmax,
                                           const float* __restrict__ psum,
                                           const float* __restrict__ psum2,
                                           float* sh, float& Z, float& Z2) {
    int t = threadIdx.x;
    float a  = pmax[t];
    float s  = psum[t];
    float s2 = psum2[t];
    float M  = blkMax(a, sh);
    float S  = blkSum(s * expf(a - M), sh);
    float S2 = blkSum(s2 * expf(2.0f * (a - M)), sh);
    Z  = M + logf(S);
    Z2 = 2.0f * M + logf(S2);
}

__device__ __forceinline__ float combineZ(const float* __restrict__ pmax,
                                          const float* __restrict__ psum, float* sh) {
    int t = threadIdx.x;
    float a = pmax[t];
    float s = psum[t];
    float M = blkMax(a, sh);
    float S = blkSum(s * expf(a - M), sh);
    return M + logf(S);
}

// Write the three log-domain partials for a block's lw values.
__device__ __forceinline__ void writePartials(float lwv, float* sh,
                                              float* pmax, float* psum, float* psum2) {
    float bm  = blkMax(lwv, sh);
    float bs  = blkSum(expf(lwv - bm), sh);
    float bs2 = blkSum(expf(2.0f * (lwv - bm)), sh);
    if (threadIdx.x == 0) {
        pmax[blockIdx.x] = bm; psum[blockIdx.x] = bs; psum2[blockIdx.x] = bs2;
    }
}

// ---------------- init: lw0 = logN(y0 | x0, I), partials, loglik = -logN ----------------
__global__ void k_init(const float* __restrict__ x0, const float* __restrict__ y,
                       float* __restrict__ lw, float* __restrict__ pmax,
                       float* __restrict__ psum, float* __restrict__ psum2,
                       float* __restrict__ sc) {
    __shared__ float sh[TPB];
    int i = blockIdx.x * TPB + threadIdx.x;
    float4 x = ((const float4*)x0)[i];
    float4 yv = *((const float4*)y);
    float d0 = yv.x - x.x, d1 = yv.y - x.y, d2 = yv.z - x.z, d3 = yv.w - x.w;
    float ll = -0.5f * (d0 * d0 + d1 * d1 + d2 * d2 + d3 * d3) - OBSC;
    lw[i] = ll;
    writePartials(ll, sh, pmax, psum, psum2);
    if (blockIdx.x == 0 && threadIdx.x == 0) sc[0] = -LOGN;   // loglik accumulator
}

// ---------------- per step 1/2: loglik accumulate + CDF only when resampling ----------------
__global__ void k_cdf(const float* __restrict__ lw, const float* __restrict__ pmax,
                      const float* __restrict__ psum, const float* __restrict__ psum2,
                      float* __restrict__ sc, float* __restrict__ cdf) {
    __shared__ float sh[TPB];
    int t = threadIdx.x;
    float Z, Z2;
    combineZZ2(pmax, psum, psum2, sh, Z, Z2);
    if (blockIdx.x == 0 && t == 0) sc[0] += Z;    // accumulate incremental loglik
    float ess = expf(2.0f * Z - Z2);              // 1 / sum(w^2), w = softmax(lw)
    if (!(ess < 0.5f * (float)NPART)) return;     // uniform across blocks

    // exclusive block offset: scan derived per-block weight sums
    float pwt = psum[t] * expf(pmax[t] - Z);
    sh[t] = pwt; __syncthreads();
    for (int off = 1; off < TPB; off <<= 1) {
        float add = (t >= off) ? sh[t - off] : 0.0f;
        __syncthreads();
        sh[t] += add;
        __syncthreads();
    }
    float boff = (blockIdx.x == 0) ? 0.0f : sh[blockIdx.x - 1];
    __syncthreads();

    int i = blockIdx.x * TPB + t;
    float w = expf(lw[i] - Z);
    sh[t] = w; __syncthreads();
    for (int off = 1; off < TPB; off <<= 1) {
        float add = (t >= off) ? sh[t - off] : 0.0f;
        __syncthreads();
        sh[t] += add;
        __syncthreads();
    }
    cdf[i] = boff + sh[t];
}

// ---------------- per step 2/2: resample-gather, transition, loglik, partials ----------------
__global__ void k_step(const float* __restrict__ xsrc, float* __restrict__ xdst,
                       float* __restrict__ lw, const float* __restrict__ eps_t,
                       const float* __restrict__ y_t, const float* __restrict__ u_t,
                       const float* __restrict__ cdf,
                       const float* __restrict__ pmaxI, const float* __restrict__ psumI,
                       const float* __restrict__ psum2I,
                       float* __restrict__ pmaxO, float* __restrict__ psumO,
                       float* __restrict__ psum2O) {
    __shared__ float sh[TPB];
    __shared__ float seps[TPB * DDIM];
    int t = threadIdx.x, b = blockIdx.x;
    const float* tile = eps_t + (size_t)b * (TPB * DDIM);

#if HAVE_TDM
    // Async Tensor-DMA of the 4KB noise tile into LDS (wave 0 issues once).
    if (t < 32) {
        u64 ga = (u64)tile;
        u32 ldsa = (u32)(u64)(&seps[0]);             // low 32 bits of generic = LDS offset
        v4u g0;
        g0.x = 1u;                                   // count = 1, user mode
        g0.y = ldsa;                                 // lds_addr
        g0.z = (u32)ga;                              // global_addr[31:0]
        g0.w = (u32)(ga >> 32) | (2u << 30);         // global_addr[56:32] | type=2
        v8i g1;
        g1[0] = (int)(2u << 16);                     // workgroup_mask=0, data_size=2 (4B)
        g1[1] = (int)((u32)(TPB * DDIM) << 16);      // tensor_dim0 = 1024 (low 16)
        g1[2] = (int)(1u << 16);                     // tensor_dim0 hi=0, tensor_dim1 = 1
        g1[3] = (int)((u32)(TPB * DDIM) << 16);      // tile_dim0 = 1024
        g1[4] = 1;                                   // tile_dim1 = 1, tile_dim2 = 0
        g1[5] = (int)(u32)(TPB * DDIM);              // tensor_dim0_stride = 1024
        g1[6] = 0;
        g1[7] = 0;
        v4i gz = {0, 0, 0, 0};
#if __clang_major__ >= 23
        v8i gz8 = {0, 0, 0, 0, 0, 0, 0, 0};
        __builtin_amdgcn_tensor_load_to_lds(g0, g1, gz, gz, gz8, 0);
#else
        __builtin_amdgcn_tensor_load_to_lds(g0, g1, gz, gz, 0);
#endif
    }
#else
    for (int k = t; k < TPB * DDIM; k += TPB) seps[k] = tile[k];
#endif

    // Overlap with the DMA: Z / ESS flag + resample index + x gather.
    float Z, Z2;
    combineZZ2(pmaxI, psumI, psum2I, sh, Z, Z2);
    bool flag = expf(2.0f * Z - Z2) < 0.5f * (float)NPART;   // matches k_cdf bitwise
    int i = b * TPB + t;

    __builtin_prefetch(xsrc + (size_t)i * DDIM, 0, 0);       // global_prefetch

    float4 xs;
    float lwb;
    if (flag) {
        float pos = (u_t[0] + (float)i) * (1.0f / (float)NPART);
        int lo = 0, hi = NPART;                              // searchsorted left
        while (lo < hi) {
            int mid = (lo + hi) >> 1;
            if (cdf[mid] < pos) lo = mid + 1; else hi = mid;
        }
        int idx = (lo < NPART) ? lo : (NPART - 1);
        xs = ((const float4*)xsrc)[idx];
        lwb = -LOGN;
    } else {
        xs = ((const float4*)xsrc)[i];
        lwb = lw[i] - Z;
    }

#if HAVE_TDM
    if (t < 32) __builtin_amdgcn_s_wait_tensorcnt(0);
#endif
    __syncthreads();   // eps tile visible to all waves

    float4 e = ((const float4*)seps)[t];
    float4 yv = *((const float4*)y_t);
    float nx0 = RHO * xs.x + SIGX * e.x;
    float nx1 = RHO * xs.y + SIGX * e.y;
    float nx2 = RHO * xs.z + SIGX * e.z;
    float nx3 = RHO * xs.w + SIGX * e.w;
    float d0 = yv.x - nx0, d1 = yv.y - nx1, d2 = yv.z - nx2, d3 = yv.w - nx3;
    float ll = -0.5f * (d0 * d0 + d1 * d1 + d2 * d2 + d3 * d3) - OBSC;
    float lwn = lwb + ll;
    ((float4*)xdst)[i] = make_float4(nx0, nx1, nx2, nx3);
    lw[i] = lwn;

    writePartials(lwn, sh, pmaxO, psumO, psum2O);
}

// ---------------- final: out[0]=loglik, out[1..N]=lw - Z ----------------
__global__ void k_final(const float* __restrict__ lw, const float* __restrict__ pmax,
                        const float* __restrict__ psum, const float* __restrict__ sc,
                        float* __restrict__ out) {
    __shared__ float sh[TPB];
    int t = threadIdx.x;
    float Z = combineZ(pmax, psum, sh);
    int i = blockIdx.x * TPB + t;
    out[1 + i] = lw[i] - Z;
    if (blockIdx.x == 0 && t == 0) out[0] = sc[0] + Z;
}

extern "C" void kernel_launch(void* const* d_in, const int* in_sizes, int n_in,
                              void* d_out, int out_size, void* d_ws, size_t ws_size,
                              hipStream_t stream) {
    const float* y  = (const float*)d_in[0];   // [256, 4]
    const float* x0 = (const float*)d_in[1];   // [65536, 4]
    const float* tn = (const float*)d_in[2];   // [255, 65536, 4]
    const float* ru = (const float*)d_in[3];   // [255]

    float* ws  = (float*)d_ws;
    float* xa    = ws;                          // N*D
    float* xb    = xa + (size_t)NPART * DDIM;   // N*D
    float* lw    = xb + (size_t)NPART * DDIM;   // N
    float* cdf   = lw + NPART;                  // N
    float* PM[2]  = { cdf + NPART,          cdf + NPART + 3 * NBLK };
    float* PS[2]  = { PM[0] + NBLK,         PM[1] + NBLK };
    float* PS2[2] = { PS[0] + NBLK,         PS[1] + NBLK };
    float* sc    = cdf + NPART + 6 * NBLK;      // [0] = loglik accumulator
    float* out   = (float*)d_out;

    dim3 grid(NBLK), block(TPB);
    // k_init writes partial set 0
    k_init<<<grid, block, 0, stream>>>(x0, y, lw, PM[0], PS[0], PS2[0], sc);

    const float* xsrc = x0;
    for (int step = 1; step < TSTEPS; ++step) {
        int pin = (step - 1) & 1, pout = step & 1;
        float* xdst = (step & 1) ? xa : xb;
        k_cdf<<<grid, block, 0, stream>>>(lw, PM[pin], PS[pin], PS2[pin], sc, cdf);
        k_step<<<grid, block, 0, stream>>>(
            xsrc, xdst, lw,
            tn + (size_t)(step - 1) * NPART * DDIM,
            y + (size_t)step * DDIM,
            ru + (step - 1),
            cdf,
            PM[pin], PS[pin], PS2[pin],
            PM[pout], PS[pout], PS2[pout]);
        xsrc = xdst;
    }
    // last written partial set is (TSTEPS-1)&1 == 1
    k_final<<<grid, block, 0, stream>>>(lw, PM[1], PS[1], sc, out);
}